// RegionDetectionRCNN_43456479101591
// MI455X (gfx1250) — compile-verified
//
#include <hip/hip_runtime.h>
#include <stdint.h>

#define BATCH       8
#define NPROP       16384
#define MCAND       (NPROP * 2)        // candidates per image (2 fg classes)
#define DET_PER_IMG 320
#define NMS_ITERS   640
#define NEGV        (-1.0e9f)
#define SCORE_TH    0.3f
#define NMS_TH      0.3f
#define BBOX_CLIP   4.135166556742356f // log(1000/16)
#define MIN_SIZE    0.01f
#define THREADS     1024
#define WAVES       (THREADS / 32)

__device__ __forceinline__ unsigned ballot32(bool p) {
#if __has_builtin(__builtin_amdgcn_ballot_w32)
  return __builtin_amdgcn_ballot_w32(p);
#else
  return (unsigned)__ballot(p);
#endif
}

__device__ __forceinline__ void amax2(float& v, int& i, float ov, int oi) {
  // argmax with lowest-index tie-break (matches jnp.argmax "first" semantics)
  if (ov > v || (ov == v && oi < i)) { v = ov; i = oi; }
}

// Decode + clip one candidate (torchvision BoxCoder), same formula everywhere.
__device__ __forceinline__ float4 decode_clip(const float* __restrict__ rel,
                                              const float* __restrict__ props,
                                              size_t gi, int lab, float fs)
{
  const float4 p = reinterpret_cast<const float4*>(props)[gi];
  const float w  = p.z - p.x;
  const float h  = p.w - p.y;
  const float cx = p.x + 0.5f * w;
  const float cy = p.y + 0.5f * h;

  const float* rp = rel + gi * 12 + (size_t)lab * 4;
  const float dx = rp[0] / 10.0f;
  const float dy = rp[1] / 10.0f;
  const float dw = fminf(rp[2] / 5.0f, BBOX_CLIP);
  const float dh = fminf(rp[3] / 5.0f, BBOX_CLIP);

  const float pcx = dx * w + cx;
  const float pcy = dy * h + cy;
  const float pw  = expf(dw) * w;
  const float ph  = expf(dh) * h;

  float4 b;
  b.x = fminf(fmaxf(pcx - 0.5f * pw, 0.0f), fs);
  b.y = fminf(fmaxf(pcy - 0.5f * ph, 0.0f), fs);
  b.z = fminf(fmaxf(pcx + 0.5f * pw, 0.0f), fs);
  b.w = fminf(fmaxf(pcy + 0.5f * ph, 0.0f), fs);
  return b;
}

// ---------------------------------------------------------------------------
// Kernel 1: fused softmax + box decode + clip + threshold + deterministic
// ordered compaction (ballot/prefix-sum, no atomics) into workspace.
// Stores the per-label OFFSET boxes (nbox = clipped + label*(fs+1)), computed
// once exactly like the reference's batched_nms, so NMS rounds need no label
// lookup. One workgroup per image; candidate order m = n*2+(label-1) kept.
// ---------------------------------------------------------------------------
extern "C" __global__ void __launch_bounds__(THREADS, 1)
rcnn_decode_compact(const float* __restrict__ logits,
                    const float* __restrict__ rel,
                    const float* __restrict__ props,
                    const int* __restrict__ img_size,
                    float4* __restrict__ wnbox,
                    float* __restrict__ wscore,
                    int* __restrict__ wcand,
                    int* __restrict__ wcount)
{
  const int img  = blockIdx.x;
  const int tid  = threadIdx.x;
  const int lane = tid & 31;
  const int wid  = tid >> 5;
  const float fs = (float)img_size[0];
  const float labOff = fs + 1.0f;

  __shared__ int sWaveCnt[WAVES];
  __shared__ int sWaveOff[WAVES];
  __shared__ int sTotal;
  __shared__ int sBase;
  if (tid == 0) sBase = 0;
  __syncthreads();

  for (int r = 0; r < MCAND / THREADS; ++r) {
    const int m    = r * THREADS + tid;
    const int n    = m >> 1;
    const int lab  = 1 + (m & 1);
    const size_t gi = (size_t)img * NPROP + n;

    const float4 b = decode_clip(rel, props, gi, lab, fs);

    // ---- softmax over 3 classes ----
    const float l0 = logits[gi * 3 + 0];
    const float l1 = logits[gi * 3 + 1];
    const float l2 = logits[gi * 3 + 2];
    const float mxl = fmaxf(l0, fmaxf(l1, l2));
    const float e0 = expf(l0 - mxl);
    const float e1 = expf(l1 - mxl);
    const float e2 = expf(l2 - mxl);
    const float sc = ((lab == 1) ? e1 : e2) / (e0 + e1 + e2);

    const bool valid = (sc > SCORE_TH) && ((b.z - b.x) >= MIN_SIZE) && ((b.w - b.y) >= MIN_SIZE);

    // ---- ordered stream compaction (wave32 ballot + block scan) ----
    const unsigned mask    = ballot32(valid);
    const int      lanePre = __popc(mask & ((1u << lane) - 1u));
    if (lane == 0) sWaveCnt[wid] = __popc(mask);
    __syncthreads();
    if (wid == 0) {
      const int v = sWaveCnt[lane];
      int x = v;
      for (int o = 1; o < 32; o <<= 1) {
        const int y = __shfl_up(x, o, 32);
        if (lane >= o) x += y;
      }
      sWaveOff[lane] = x - v;
      if (lane == 31) sTotal = x;
    }
    __syncthreads();
    if (valid) {
      const size_t o = (size_t)img * MCAND + (size_t)(sBase + sWaveOff[wid] + lanePre);
      const float so = (float)lab * labOff;             // nbox = box + off (one rounding,
      wnbox[o]  = make_float4(b.x + so, b.y + so,       //  exactly like the reference)
                              b.z + so, b.w + so);
      wscore[o] = sc;
      wcand[o]  = m;
    }
    __syncthreads();
    if (tid == 0) sBase += sTotal;
    __syncthreads();
  }
  if (tid == 0) wcount[img] = sBase;
}

// ---------------------------------------------------------------------------
// Kernel 2: greedy NMS with the full score array resident in LDS (128 KB of
// the WGP's 320 KB), staged via the Tensor Data Mover; then label-2-first
// stable reorder + truncation to 320; output boxes re-decoded (un-offset).
// ---------------------------------------------------------------------------
extern "C" __global__ void __launch_bounds__(THREADS, 1)
rcnn_nms_out(const float4* __restrict__ wnbox,
             const float* __restrict__ wscore,
             const int* __restrict__ wcand,
             const int* __restrict__ wcount,
             const float* __restrict__ rel,
             const float* __restrict__ props,
             const int* __restrict__ img_size,
             float* __restrict__ out)
{
  const int img  = blockIdx.x;
  const int tid  = threadIdx.x;
  const int lane = tid & 31;
  const int wid  = tid >> 5;
  const float fs = (float)img_size[0];

  __shared__ float sScore[MCAND];        // 128 KB: full per-image score array
  __shared__ int   sSelIdx[NMS_ITERS];
  __shared__ float sSelScore[NMS_ITERS];
  __shared__ int   sSelLab[NMS_ITERS];
  __shared__ float sRedV[WAVES];
  __shared__ int   sRedI[WAVES];
  __shared__ float sBestV;
  __shared__ int   sBestI;
  __shared__ int   sCnt;
  __shared__ int   sN2;

  const int V = wcount[img];
  const float4* boxB = wnbox  + (size_t)img * MCAND;   // offset boxes
  const float*  scB  = wscore + (size_t)img * MCAND;
  const int*    cdB  = wcand  + (size_t)img * MCAND;

  float* oBox = out + (size_t)img * DET_PER_IMG * 4;
  float* oSc  = out + (size_t)BATCH * DET_PER_IMG * 4 + (size_t)img * DET_PER_IMG;
  float* oLb  = out + (size_t)BATCH * DET_PER_IMG * 5 + (size_t)img * DET_PER_IMG;
  for (int k = tid; k < DET_PER_IMG * 4; k += THREADS) oBox[k] = 0.0f;
  for (int k = tid; k < DET_PER_IMG; k += THREADS) { oSc[k] = 0.0f; oLb[k] = 0.0f; }
  if (tid == 0) sCnt = 0;

  // warm L2 with the box records reused every NMS round (global_prefetch_b8)
  for (int j = tid * 8; j < V; j += THREADS * 8) __builtin_prefetch(&boxB[j], 0, 1);

  // ---- stage compacted scores into LDS (TDM path on gfx1250) ----
#if defined(__gfx1250__) && __has_builtin(__builtin_amdgcn_tensor_load_to_lds) && __has_builtin(__builtin_amdgcn_s_wait_tensorcnt)
  if (tid < 32) {                         // a single wave issues one DMA
    if (V > 0) {
      typedef unsigned int v4u __attribute__((ext_vector_type(4)));
      typedef int v8i __attribute__((ext_vector_type(8)));
      typedef int v4i __attribute__((ext_vector_type(4)));
      const unsigned long long ga = (unsigned long long)(uintptr_t)scB;
      const unsigned lds = (unsigned)(uintptr_t)&sScore[0];
      v4u g0;
      g0[0] = 1u;                                              // count=1, user desc
      g0[1] = lds;                                             // lds_addr
      g0[2] = (unsigned)(ga & 0xffffffffull);                  // global_addr lo
      g0[3] = (unsigned)((ga >> 32) & 0x01ffffffull) | (2u << 30); // addr hi | type=2
      v8i g1;
      g1[0] = (int)(2u << 16);                                 // data_size = 4B
      g1[1] = (int)(((unsigned)V & 0xffffu) << 16);            // tensor_dim0 lo16
      g1[2] = (int)((((unsigned)V >> 16) & 0xffffu) | (1u << 16)); // dim0 hi | tensor_dim1=1
      g1[3] = (int)(((unsigned)V & 0xffffu) << 16);            // tile_dim0 = V
      g1[4] = 1;                                               // tile_dim1 = 1
      g1[5] = V;                                               // tensor_dim0_stride
      g1[6] = 0;
      g1[7] = 0;
      const v4i z4 = {0, 0, 0, 0};
      const v8i z8 = {0, 0, 0, 0, 0, 0, 0, 0};
      __builtin_amdgcn_tensor_load_to_lds(g0, g1, z4, z4, z8, 0);
    }
    __builtin_amdgcn_s_wait_tensorcnt((short)0);
  }
#else
  for (int j = tid; j < V; j += THREADS) sScore[j] = scB[j];
#endif
  __syncthreads();

  // ---- greedy NMS: argmax (LDS) -> suppress ----
  for (int it = 0; it < NMS_ITERS; ++it) {
    float bv = -3.0e38f;
    int   bi = 0x7fffffff;
    for (int j = tid; j < V; j += THREADS) {   // ascending j => strict > keeps lowest idx
      const float s = sScore[j];
      if (s > bv) { bv = s; bi = j; }
    }
    for (int o = 16; o > 0; o >>= 1)
      amax2(bv, bi, __shfl_xor(bv, o, 32), __shfl_xor(bi, o, 32));
    if (lane == 0) { sRedV[wid] = bv; sRedI[wid] = bi; }
    __syncthreads();
    if (wid == 0) {
      bv = sRedV[lane]; bi = sRedI[lane];
      for (int o = 16; o > 0; o >>= 1)
        amax2(bv, bi, __shfl_xor(bv, o, 32), __shfl_xor(bi, o, 32));
      if (lane == 0) { sBestV = bv; sBestI = bi; }
    }
    __syncthreads();
    const float bestV = sBestV;
    const int   bestI = sBestI;
    if (!(bestV > 0.5f * NEGV)) break;      // no live candidates left: rest are ok=false

    if (tid == 0) {
      sSelIdx[sCnt]   = bestI;
      sSelScore[sCnt] = bestV;
      sSelLab[sCnt]   = 1 + (cdB[bestI] & 1);
      sCnt            = sCnt + 1;
    }
    const float4 a = boxB[bestI];           // already offset, as in the reference
    const float aArea = (a.z - a.x) * (a.w - a.y);
    for (int j = tid; j < V; j += THREADS) {
      if (j == bestI) { sScore[j] = NEGV; continue; }
      if (!(sScore[j] > 0.5f * NEGV)) continue;   // already dead: masking is a no-op
      const float4 b = boxB[j];
      const float iw = fmaxf(fminf(a.z, b.z) - fmaxf(a.x, b.x), 0.0f);
      const float ih = fmaxf(fminf(a.w, b.w) - fmaxf(a.y, b.y), 0.0f);
      const float inter = iw * ih;
      const float bArea = (b.z - b.x) * (b.w - b.y);
      const float iou = inter / (aArea + bArea - inter);   // NaN>th is false, matches jnp
      if (iou > NMS_TH) sScore[j] = NEGV;
    }
    __syncthreads();
  }
  __syncthreads();

  // ---- stable partition: label==2 first, then label==1; truncate to 320 ----
  const int cnt = sCnt;
  if (tid == 0) {
    int n2 = 0;
    for (int j = 0; j < cnt; ++j) n2 += (sSelLab[j] == 2);
    sN2 = n2;
  }
  __syncthreads();
  for (int k = tid; k < cnt; k += THREADS) {
    const int lab = sSelLab[k];
    int pre = 0;
    for (int j = 0; j < k; ++j) pre += (sSelLab[j] == lab);
    const int pos = (lab == 2) ? pre : (sN2 + pre);
    if (pos < DET_PER_IMG) {
      // re-decode the un-offset clipped box for output (<=640 per image)
      const int m = cdB[sSelIdx[k]];
      const float4 b = decode_clip(rel, props, (size_t)img * NPROP + (m >> 1), 1 + (m & 1), fs);
      oBox[pos * 4 + 0] = b.x;
      oBox[pos * 4 + 1] = b.y;
      oBox[pos * 4 + 2] = b.z;
      oBox[pos * 4 + 3] = b.w;
      oSc[pos] = sSelScore[k];
      oLb[pos] = (float)lab;
    }
  }
}

// ---------------------------------------------------------------------------
extern "C" void kernel_launch(void* const* d_in, const int* in_sizes, int n_in,
                              void* d_out, int out_size, void* d_ws, size_t ws_size,
                              hipStream_t stream)
{
  (void)in_sizes; (void)n_in; (void)out_size; (void)ws_size;
  const float* logits = (const float*)d_in[0];
  const float* rel    = (const float*)d_in[1];
  const float* props  = (const float*)d_in[2];
  const int*   imgsz  = (const int*)d_in[3];

  char* ws = (char*)d_ws;                                   // ~6.3 MB used
  float4* wnbox  = (float4*)(ws);                           // B*M*16 (offset boxes)
  float*  wscore = (float*)(ws + (size_t)BATCH * MCAND * 16);
  int*    wcand  = (int*)(ws + (size_t)BATCH * MCAND * 20);
  int*    wcount = (int*)(ws + (size_t)BATCH * MCAND * 24);

  rcnn_decode_compact<<<dim3(BATCH), dim3(THREADS), 0, stream>>>(
      logits, rel, props, imgsz, wnbox, wscore, wcand, wcount);
  rcnn_nms_out<<<dim3(BATCH), dim3(THREADS), 0, stream>>>(
      wnbox, wscore, wcand, wcount, rel, props, imgsz, (float*)d_out);
}